// EncodingLayer_85031762526316
// MI455X (gfx1250) — compile-verified
//
#include <hip/hip_runtime.h>

#define N_ 64
#define T_ 4096
#define K_ 32
#define D_ 128
#define NW 8          // waves per block
#define CHUNKS 8      // blocks per batch item
#define ROWS_PER_CHUNK (T_ / CHUNKS)               // 512
#define ITERS (ROWS_PER_CHUNK / (NW * 16))         // 4
#define XROWB ((D_ + 4) * 4)                       // padded LDS row stride: 528B

typedef __attribute__((ext_vector_type(2))) float v2f;
typedef __attribute__((ext_vector_type(8))) float v8f;

// Async global->LDS fill of one 16x128 fp32 tile (16B per lane per row).
// Tracked by ASYNCcnt; completes in order.
__device__ __forceinline__ void issue_x_tile(const float* gp, uint32_t ldsb) {
#pragma unroll
    for (int r = 0; r < 16; ++r) {
        asm volatile("global_load_async_to_lds_b128 %0, %1, off"
                     :: "v"(ldsb + (uint32_t)(r * XROWB)),
                        "v"((unsigned long long)(uintptr_t)(gp + r * D_))
                     : "memory");
    }
}

// Fused: cross-GEMM (fp32 WMMA) -> softmax over K -> A^T*x GEMM (fp32 WMMA),
// x tiles streamed global->LDS with double-buffered async copies.
__global__ void __launch_bounds__(256) enc_main(const float* __restrict__ xg,
                                                const float* __restrict__ cwg,
                                                const float* __restrict__ scg,
                                                float* __restrict__ wsE,
                                                float* __restrict__ wsS) {
    __shared__ float cw[K_][D_ + 4];          // codewords
    __shared__ float xs[NW][2][16][D_ + 4];   // per-wave double-buffered x tile
    __shared__ float at[NW][K_][18];          // per-wave A^T staging (72B rows)
    __shared__ float ered[K_][D_];            // block E reduction buffer
    __shared__ float sred[K_];                // block S reduction buffer
    __shared__ float sqxs[NW][16];            // per-wave ||x_t||^2
    __shared__ float scl[K_], s2l[K_], scc[K_];

    const int tid  = threadIdx.x;
    const int w    = tid >> 5;
    const int lane = tid & 31;
    const int lh   = lane >> 4;   // lane half (0/1)
    const int ll   = lane & 15;   // lane within half
    const int n     = blockIdx.y;
    const int chunk = blockIdx.x;

    // ---- block init: codewords to LDS, zero reduction buffers ----
    for (int i = tid; i < K_ * D_; i += 256) cw[i >> 7][i & 127] = cwg[i];
    for (int i = tid; i < K_ * D_; i += 256) ered[i >> 7][i & 127] = 0.0f;
    if (tid < K_) sred[tid] = 0.0f;
    __syncthreads();
    if (tid < K_) {
        float s = 0.0f;
        for (int d = 0; d < D_; ++d) s += cw[tid][d] * cw[tid][d];
        float sc = scg[tid];
        scl[tid] = sc;               // scale[k]
        s2l[tid] = -2.0f * sc;       // -2*scale[k]
        scc[tid] = sc * s;           // scale[k]*||c_k||^2
    }
    __syncthreads();

    const float sk0 = scl[ll],      sk1 = scl[ll + 16];
    const float s20 = s2l[ll],      s21 = s2l[ll + 16];
    const float sc0 = scc[ll],      sc1 = scc[ll + 16];

    v8f acc0[8], acc1[8];           // E accum: 32 (k) x 128 (d) fp32 per wave
    for (int i = 0; i < 8; ++i) { v8f z = {}; acc0[i] = z; acc1[i] = z; }
    float sA0 = 0.0f, sA1 = 0.0f;   // column sums of A

    // per-wave LDS byte offsets of the two x-tile buffers (+ lane's 16B slot)
    const uint32_t xsoff0 =
        (uint32_t)(uintptr_t)&xs[w][0][0][0] + (uint32_t)(lane * 16);
    const uint32_t xsoff1 =
        (uint32_t)(uintptr_t)&xs[w][1][0][0] + (uint32_t)(lane * 16);
    const float* xlane = xg + (size_t)n * T_ * D_ + lane * 4;

    // prologue: start async fill of tile 0 into buffer 0
    {
        const int row0 = chunk * ROWS_PER_CHUNK + w * 16;
        issue_x_tile(xlane + (size_t)row0 * D_, xsoff0);
    }

    for (int it = 0; it < ITERS; ++it) {
        const int buf = it & 1;

        if (it + 1 < ITERS) {
            // kick off next tile into the other buffer, then wait for current
            const int rown = chunk * ROWS_PER_CHUNK + ((it + 1) * NW + w) * 16;
            issue_x_tile(xlane + (size_t)rown * D_, buf ? xsoff0 : xsoff1);
            asm volatile("s_wait_asynccnt 16" ::: "memory");
        } else {
            asm volatile("s_wait_asynccnt 0" ::: "memory");
        }

        // ---- ||x_t||^2 : 2 lanes per row, cross-half combine ----
        {
            const float4* p = (const float4*)&xs[w][buf][ll][lh * 64];
            float s = 0.0f;
            for (int i = 0; i < 16; ++i) {
                float4 v = p[i];
                s += v.x * v.x + v.y * v.y + v.z * v.z + v.w * v.w;
            }
            s += __shfl_xor(s, 16, 32);
            if (lh == 0) sqxs[w][ll] = s;
        }

        // ---- GEMM1: cross[16t x 32k] = x_tile . c^T  (fp32 WMMA) ----
        v8f c0 = {}, c1 = {};
        for (int d0 = 0; d0 < D_; d0 += 4) {
            v2f a  = *(const v2f*)&xs[w][buf][ll][d0 + 2 * lh];
            v2f b0 = *(const v2f*)&cw[ll][d0 + 2 * lh];
            v2f b1 = *(const v2f*)&cw[ll + 16][d0 + 2 * lh];
            c0 = __builtin_amdgcn_wmma_f32_16x16x4_f32(false, a, false, b0, (short)0, c0, false, false);
            c1 = __builtin_amdgcn_wmma_f32_16x16x4_f32(false, a, false, b1, (short)0, c1, false, false);
        }

        // ---- softmax over K=32 (each row lives in one 16-lane half x 2 regs) ----
        for (int r = 0; r < 8; ++r) {
            float q  = sqxs[w][r + 8 * lh];
            float v0 = fmaf(s20, c0[r], fmaf(sk0, q, sc0));
            float v1 = fmaf(s21, c1[r], fmaf(sk1, q, sc1));
            float m = fmaxf(v0, v1);
            m = fmaxf(m, __shfl_xor(m, 1, 32));
            m = fmaxf(m, __shfl_xor(m, 2, 32));
            m = fmaxf(m, __shfl_xor(m, 4, 32));
            m = fmaxf(m, __shfl_xor(m, 8, 32));
            float e0 = __expf(v0 - m), e1 = __expf(v1 - m);
            float s = e0 + e1;
            s += __shfl_xor(s, 1, 32);
            s += __shfl_xor(s, 2, 32);
            s += __shfl_xor(s, 4, 32);
            s += __shfl_xor(s, 8, 32);
            float inv = 1.0f / s;
            float a0 = e0 * inv, a1 = e1 * inv;
            sA0 += a0; sA1 += a1;
            at[w][ll][r + 8 * lh]      = a0;   // A^T staging: [k][t]
            at[w][ll + 16][r + 8 * lh] = a1;
        }

        // ---- GEMM2: E[32k x 128d] += A^T . x_tile  (fp32 WMMA) ----
        for (int s4 = 0; s4 < 16; s4 += 4) {
            const int tr = s4 + 2 * lh;
            v2f aa0 = *(const v2f*)&at[w][ll][tr];
            v2f aa1 = *(const v2f*)&at[w][ll + 16][tr];
            for (int dt = 0; dt < 8; ++dt) {
                v2f b;
                b.x = xs[w][buf][tr][dt * 16 + ll];
                b.y = xs[w][buf][tr + 1][dt * 16 + ll];
                acc0[dt] = __builtin_amdgcn_wmma_f32_16x16x4_f32(false, aa0, false, b, (short)0, acc0[dt], false, false);
                acc1[dt] = __builtin_amdgcn_wmma_f32_16x16x4_f32(false, aa1, false, b, (short)0, acc1[dt], false, false);
            }
        }
    }

    // ---- block reduction: S ----
    float t0 = sA0 + __shfl_xor(sA0, 16, 32);
    float t1 = sA1 + __shfl_xor(sA1, 16, 32);
    if (lh == 0) {
        atomicAdd(&sred[ll], t0);
        atomicAdd(&sred[ll + 16], t1);
    }
    // ---- block reduction: E (ds_add_f32) ----
    for (int dt = 0; dt < 8; ++dt)
        for (int r = 0; r < 8; ++r) {
            atomicAdd(&ered[r + 8 * lh][dt * 16 + ll], acc0[dt][r]);
            atomicAdd(&ered[16 + r + 8 * lh][dt * 16 + ll], acc1[dt][r]);
        }
    __syncthreads();

    // ---- flush block partials to workspace (global fp32 atomics) ----
    float* wsEn = wsE + (size_t)n * (K_ * D_);
    for (int i = tid; i < K_ * D_; i += 256)
        atomicAdd(&wsEn[i], ered[i >> 7][i & 127]);
    if (tid < K_) atomicAdd(&wsS[n * K_ + tid], sred[tid]);
}

// E[n,k,d] = accE[n,k,d] - S[n,k]*c[k,d]
__global__ void __launch_bounds__(256) enc_final(const float* __restrict__ wsE,
                                                 const float* __restrict__ wsS,
                                                 const float* __restrict__ cwg,
                                                 float* __restrict__ out) {
    int i = blockIdx.x * 256 + threadIdx.x;
    if (i >= N_ * K_ * D_) return;
    int kd = i & (K_ * D_ - 1);
    int n  = i >> 12;                 // K_*D_ = 4096
    out[i] = wsE[i] - wsS[n * K_ + (kd >> 7)] * cwg[kd];
}

extern "C" void kernel_launch(void* const* d_in, const int* in_sizes, int n_in,
                              void* d_out, int out_size, void* d_ws, size_t ws_size,
                              hipStream_t stream) {
    (void)in_sizes; (void)n_in; (void)out_size; (void)ws_size;
    const float* x  = (const float*)d_in[0];
    const float* cw = (const float*)d_in[1];
    const float* sc = (const float*)d_in[2];

    float* wsE = (float*)d_ws;
    float* wsS = wsE + (size_t)N_ * K_ * D_;

    hipMemsetAsync(d_ws, 0, ((size_t)N_ * K_ * D_ + (size_t)N_ * K_) * sizeof(float), stream);

    dim3 g1(CHUNKS, N_);
    enc_main<<<g1, 256, 0, stream>>>(x, cw, sc, wsE, wsS);

    const int total = N_ * K_ * D_;
    enc_final<<<(total + 255) / 256, 256, 0, stream>>>(wsE, wsS, cw, (float*)d_out);
}